// TtLlamaAttention_63307817943116
// MI455X (gfx1250) — compile-verified
//
#include <hip/hip_runtime.h>

// ---------------------------------------------------------------------------
// Llama decode attention step for MI455X (gfx1250, wave32).
// Memory-bound (~235 MB weight+cache traffic -> ~10us @ 23.3 TB/s):
//  * big GEMVs use V_WMMA_F32_16X16X4_F32, split-K x16, software-pipelined,
//    immediate-offset B loads (no per-iter address math), zero-padded A
//  * attention: split-KV flash decode, streaming NT reads of K/V cache
// ---------------------------------------------------------------------------

#define BB      4
#define NH      32
#define NKVH    8
#define HD      128
#define HIDDEN  4096
#define WINSZ   8192
#define NQKV    6144          // (NH + 2*NKVH) * HD
#define GRP     4             // NH / NKVH
#define NPART   4             // split-KV partitions for attention
#define NSPLIT  16            // split-K for the big GEMVs
#define ASCALE  0.08838834764831845f   // D^-0.5

typedef __attribute__((ext_vector_type(2))) float v2f;
typedef __attribute__((ext_vector_type(4))) float v4f;
typedef __attribute__((ext_vector_type(8))) float v8f;

#define WMMA4(a, b, c) \
  __builtin_amdgcn_wmma_f32_16x16x4_f32(false, (a), false, (b), (short)0, (c), false, false)

#define NTLOADF(p)  __builtin_nontemporal_load(p)

// Zero-pad A into a dense 16 x 4096 buffer (rows >= M are zero) so the GEMM
// A-fragment loads are unconditional v2f loads with no exec-mask divergence.
__global__ void pad_a16_4096(const float* __restrict__ src,
                             float* __restrict__ dst, int M) {
  const int i = blockIdx.x * blockDim.x + threadIdx.x;   // 16*4096
  if (i >= 16 * HIDDEN) return;
  const int m = i >> 12, kk = i & 4095;
  dst[i] = (m < M) ? src[m * HIDDEN + kk] : 0.0f;
}

// ---------------------------------------------------------------------------
// C[M x N] = A16[16 x K] * B[K x N], split-K across gridDim.y slices.
// One wave per 16-column tile per K-slice. Fragments per ISA 7.12.2:
//  A: lane L -> m = L&15, khalf = L>>4; v2f = A[m][k+2*khalf .. +1]
//  B: lane L -> n = L&15;   rows k+2*khalf, k+2*khalf+1 (mirror of A)
//  C: lane L holds column n = L&15; VGPR r is row (L<16 ? r : 8+r).
// Addressing: per-lane base pointers bumped by 8 (A) / 8*N (B) per step;
// the four B row loads use immediate offsets 0, N, 4N, 5N (elements).
// 2-stage software pipeline overlaps the next fragment fetch with WMMA.
// ---------------------------------------------------------------------------
__global__ void gemm_a16_splitk(const float* __restrict__ A16,
                                const float* __restrict__ B,
                                float* __restrict__ Cpart,
                                int M, int K, int N) {
  const int lane  = threadIdx.x & 31;
  const int tile  = blockIdx.x * (blockDim.x >> 5) + (threadIdx.x >> 5);
  if (tile * 16 >= N) return;
  const int m     = lane & 15;
  const int khalf = lane >> 4;
  const int col   = tile * 16 + m;
  const int kchunk = K / gridDim.y;
  const int k0     = blockIdx.y * kchunk;

  const float* Ap = A16 + (size_t)m * K + k0 + 2 * khalf;       // 8B aligned
  const float* Bp = B + (size_t)(k0 + 2 * khalf) * N + col;
  const size_t bStep = (size_t)8 * N;
  const size_t offN1 = (size_t)N;
  const size_t offN4 = (size_t)4 * N;
  const size_t offN5 = (size_t)5 * N;

  v8f acc  = {};
  v8f acc2 = {};

  // pipeline prologue: fragment for k-step 0
  v2f a0 = *(const v2f*)(Ap);
  v2f a1 = *(const v2f*)(Ap + 4);
  v2f b0, b1;
  b0.x = NTLOADF(Bp);
  b0.y = NTLOADF(Bp + offN1);
  b1.x = NTLOADF(Bp + offN4);
  b1.y = NTLOADF(Bp + offN5);

  for (int k = 8; k < kchunk; k += 8) {
    Ap += 8;
    Bp += bStep;
    const v2f na0 = *(const v2f*)(Ap);
    const v2f na1 = *(const v2f*)(Ap + 4);
    v2f nb0, nb1;
    nb0.x = NTLOADF(Bp);
    nb0.y = NTLOADF(Bp + offN1);
    nb1.x = NTLOADF(Bp + offN4);
    nb1.y = NTLOADF(Bp + offN5);
    acc  = WMMA4(a0, b0, acc);
    acc2 = WMMA4(a1, b1, acc2);
    a0 = na0; a1 = na1; b0 = nb0; b1 = nb1;
  }
  acc  = WMMA4(a0, b0, acc);
  acc2 = WMMA4(a1, b1, acc2);

  float* Cp = Cpart + (size_t)blockIdx.y * M * N;
#pragma unroll
  for (int r = 0; r < 8; ++r) {
    const int mm = khalf * 8 + r;
    if (mm < M) Cp[(size_t)mm * N + col] = acc[r] + acc2[r];
  }
}

// Sum split-K partials: out[i] = sum_y part[y*n + i]
__global__ void reduce_slices(const float* __restrict__ part,
                              float* __restrict__ out, int n, int nsplit) {
  const int i = blockIdx.x * blockDim.x + threadIdx.x;
  if (i >= n) return;
  float s = 0.0f;
  for (int y = 0; y < nsplit; ++y) s += part[(size_t)y * n + i];
  out[i] = s;
}

// ---------------------------------------------------------------------------
// Rotary: [q rows (128 = B*H); k rows (32 = B*KVH)] @ rot[128x128] via WMMA.
// 160 rows -> 10 M-tiles x 8 N-tiles = 80 wave-tiles. Gathers A rows out of
// the packed qkv buffer, writes a dense 160x128 "qkrot" buffer.
// ---------------------------------------------------------------------------
__global__ void rotary_wmma(const float* __restrict__ qkv,
                            const float* __restrict__ rot,
                            float* __restrict__ qkrot) {
  const int lane = threadIdx.x & 31;
  const int tile = blockIdx.x * (blockDim.x >> 5) + (threadIdx.x >> 5);
  if (tile >= 80) return;
  const int tm = tile >> 3;          // 0..9
  const int tn = tile & 7;           // 0..7
  const int khalf = lane >> 4;
  const int mrow = tm * 16 + (lane & 15);
  int srcoff;
  if (mrow < BB * NH) {              // q rows
    const int b = mrow >> 5, h = mrow & 31;
    srcoff = b * NQKV + h * HD;
  } else {                           // k rows
    const int i = mrow - BB * NH;
    const int b = i >> 3;
    srcoff = b * NQKV + NH * HD + (i & 7) * HD;
  }
  const float* Ap = qkv + srcoff + 2 * khalf;        // 8B aligned
  const float* Bp = rot + 2 * khalf * HD + tn * 16 + (lane & 15);

  v8f acc  = {};
  v8f acc2 = {};
#pragma unroll
  for (int k = 0; k < HD; k += 8) {
    const v2f a0 = *(const v2f*)(Ap + k);
    const v2f a1 = *(const v2f*)(Ap + k + 4);
    v2f b0, b1;
    const float* Bk = Bp + k * HD;
    b0.x = Bk[0];
    b0.y = Bk[HD];
    b1.x = Bk[4 * HD];
    b1.y = Bk[5 * HD];
    acc  = WMMA4(a0, b0, acc);
    acc2 = WMMA4(a1, b1, acc2);
  }
#pragma unroll
  for (int r = 0; r < 8; ++r) {
    const int mm = tm * 16 + khalf * 8 + r;
    qkrot[(size_t)mm * HD + tn * 16 + (lane & 15)] = acc[r] + acc2[r];
  }
}

// ------------------------- wave32 reductions -------------------------------
__device__ __forceinline__ float waveMax(float v) {
#pragma unroll
  for (int o = 16; o > 0; o >>= 1) v = fmaxf(v, __shfl_xor(v, o, 32));
  return v;
}
__device__ __forceinline__ float waveSum(float v) {
#pragma unroll
  for (int o = 16; o > 0; o >>= 1) v += __shfl_xor(v, o, 32);
  return v;
}

// ---------------------------------------------------------------------------
// Split-KV flash decode. grid = (B*KVH, NPART), 256 threads.
// Each block: 4 query heads (one KV group), its partition of s in [0, pos].
// Scores: thread t handles position s0+t (contiguous v4f NT K reads).
// AV: thread t owns (g, d) pairs, V row read coalesced+broadcast (NT).
// Fresh k/v at s == pos substituted in-register (inputs never mutated).
// Partial layout per (bk, part): [m[4] | l[4] | acc[4][128]] = 520 floats.
// ---------------------------------------------------------------------------
__global__ void attn_flash_kernel(const float* __restrict__ qkrot,
                                  const float* __restrict__ qkv,
                                  const float* __restrict__ cache_k,
                                  const float* __restrict__ cache_v,
                                  const int* __restrict__ pos_p,
                                  float* __restrict__ partials) {
  const int bk   = blockIdx.x;               // b*8 + kvh
  const int part = blockIdx.y;
  const int b    = bk >> 3;
  const int kvh  = bk & 7;
  const int tid  = threadIdx.x;              // 256
  const int pos  = pos_p[0];
  const int S    = pos + 1;
  const int Cs   = (((S + NPART - 1) / NPART) + 255) & ~255;  // mult of 256
  const int sBeg = part * Cs;
  const int sEnd = (S < sBeg + Cs) ? S : (sBeg + Cs);

  __shared__ __align__(16) float q_s[4 * HD];
  __shared__ __align__(16) float kf[HD];
  __shared__ __align__(16) float vf[HD];
  __shared__ float p_s[4 * 256];
  __shared__ float redbuf[8 * 4];
  __shared__ float m_sh[4], l_sh[4], rs_sh[4];

  for (int i = tid; i < 4 * HD; i += 256) {
    const int g = i >> 7, dd = i & 127;
    q_s[i] = qkrot[(size_t)(b * NH + kvh * GRP + g) * HD + dd] * ASCALE;
  }
  if (tid < HD) {
    kf[tid] = qkrot[(size_t)(BB * NH + bk) * HD + tid];
    vf[tid] = qkv[(size_t)b * NQKV + (NH + NKVH) * HD + kvh * HD + tid];
  }
  if (tid < 4) { m_sh[tid] = -3.0e38f; l_sh[tid] = 0.0f; }
  __syncthreads();

  float acc0 = 0.0f, acc1 = 0.0f;
  const int g0 = tid >> 7, g1 = g0 + 2, d = tid & 127;
  const int wid = tid >> 5, lane = tid & 31;
  const float* Kc = cache_k + (size_t)bk * WINSZ * HD;
  const float* Vc = cache_v + (size_t)bk * WINSZ * HD;
  const v4f* q0v = (const v4f*)(q_s);
  const v4f* q1v = (const v4f*)(q_s + HD);
  const v4f* q2v = (const v4f*)(q_s + 2 * HD);
  const v4f* q3v = (const v4f*)(q_s + 3 * HD);

  for (int s0 = sBeg; s0 < sEnd; s0 += 256) {
    const int rows = ((sEnd - s0) < 256) ? (sEnd - s0) : 256;
    const int s = s0 + tid;
    float e0 = -3.0e38f, e1 = -3.0e38f, e2 = -3.0e38f, e3 = -3.0e38f;
    if (tid < rows) {
      e0 = e1 = e2 = e3 = 0.0f;
      if (s == pos) {
        const v4f* kr = (const v4f*)kf;
#pragma unroll 8
        for (int dd = 0; dd < HD / 4; ++dd) {
          const v4f kv = kr[dd];
          const v4f a0 = q0v[dd], a1 = q1v[dd], a2 = q2v[dd], a3 = q3v[dd];
          e0 += a0.x * kv.x + a0.y * kv.y + a0.z * kv.z + a0.w * kv.w;
          e1 += a1.x * kv.x + a1.y * kv.y + a1.z * kv.z + a1.w * kv.w;
          e2 += a2.x * kv.x + a2.y * kv.y + a2.z * kv.z + a2.w * kv.w;
          e3 += a3.x * kv.x + a3.y * kv.y + a3.z * kv.z + a3.w * kv.w;
        }
      } else {
        const v4f* kr = (const v4f*)(Kc + (size_t)s * HD);
#pragma unroll 8
        for (int dd = 0; dd < HD / 4; ++dd) {
          const v4f kv = NTLOADF(kr + dd);
          const v4f a0 = q0v[dd], a1 = q1v[dd], a2 = q2v[dd], a3 = q3v[dd];
          e0 += a0.x * kv.x + a0.y * kv.y + a0.z * kv.z + a0.w * kv.w;
          e1 += a1.x * kv.x + a1.y * kv.y + a1.z * kv.z + a1.w * kv.w;
          e2 += a2.x * kv.x + a2.y * kv.y + a2.z * kv.z + a2.w * kv.w;
          e3 += a3.x * kv.x + a3.y * kv.y + a3.z * kv.z + a3.w * kv.w;
        }
      }
    }
    // block max per g (wave shuffle + cross-wave LDS)
    const float w0 = waveMax(e0), w1 = waveMax(e1), w2 = waveMax(e2), w3 = waveMax(e3);
    if (lane == 0) {
      redbuf[wid * 4 + 0] = w0; redbuf[wid * 4 + 1] = w1;
      redbuf[wid * 4 + 2] = w2; redbuf[wid * 4 + 3] = w3;
    }
    __syncthreads();                                         // A
    if (tid < 4) {
      float mx = redbuf[tid];
      for (int w = 1; w < 8; ++w) mx = fmaxf(mx, redbuf[w * 4 + tid]);
      const float mn = fmaxf(m_sh[tid], mx);
      rs_sh[tid] = __expf(m_sh[tid] - mn);
      m_sh[tid] = mn;
    }
    __syncthreads();                                         // B
    const float p0 = __expf(e0 - m_sh[0]);
    const float p1 = __expf(e1 - m_sh[1]);
    const float p2 = __expf(e2 - m_sh[2]);
    const float p3 = __expf(e3 - m_sh[3]);
    p_s[0 * 256 + tid] = p0; p_s[1 * 256 + tid] = p1;
    p_s[2 * 256 + tid] = p2; p_s[3 * 256 + tid] = p3;
    const float z0 = waveSum(p0), z1 = waveSum(p1), z2 = waveSum(p2), z3 = waveSum(p3);
    if (lane == 0) {
      redbuf[wid * 4 + 0] = z0; redbuf[wid * 4 + 1] = z1;
      redbuf[wid * 4 + 2] = z2; redbuf[wid * 4 + 3] = z3;
    }
    __syncthreads();                                         // C
    if (tid < 4) {
      float sm = 0.0f;
      for (int w = 0; w < 8; ++w) sm += redbuf[w * 4 + tid];
      l_sh[tid] = l_sh[tid] * rs_sh[tid] + sm;
    }
    __syncthreads();                                         // D
    // rescale accumulators, then AV with coalesced-broadcast V row reads
    acc0 *= rs_sh[g0];
    acc1 *= rs_sh[g1];
    for (int j = 0; j < rows; ++j) {
      const int sj = s0 + j;
      float vv = NTLOADF(Vc + (size_t)sj * HD + d);
      if (sj == pos) vv = vf[d];
      acc0 = fmaf(p_s[g0 * 256 + j], vv, acc0);
      acc1 = fmaf(p_s[g1 * 256 + j], vv, acc1);
    }
    __syncthreads();                                         // E
  }

  float* P = partials + (size_t)(bk * NPART + part) * 520;
  if (tid < 4) { P[tid] = m_sh[tid]; P[4 + tid] = l_sh[tid]; }
  P[8 + g0 * HD + d] = acc0;
  P[8 + g1 * HD + d] = acc1;
}

// Log-sum-exp merge of the NPART split-KV partials -> attn[B*HIDDEN]
__global__ void attn_combine_kernel(const float* __restrict__ partials,
                                    float* __restrict__ attn) {
  const int i = blockIdx.x * blockDim.x + threadIdx.x;
  if (i >= BB * HIDDEN) return;
  const int b = i >> 12, rem = i & 4095;
  const int h = rem >> 7, d = rem & 127;
  const int kvh = h >> 2, g = h & 3;
  const int bk = b * NKVH + kvh;
  const float* P = partials + (size_t)bk * NPART * 520;
  float M = -3.0e38f;
  for (int p = 0; p < NPART; ++p) M = fmaxf(M, P[p * 520 + g]);
  float num = 0.0f, den = 0.0f;
  for (int p = 0; p < NPART; ++p) {
    const float w = __expf(P[p * 520 + g] - M);
    den += w * P[p * 520 + 4 + g];
    num += w * P[(size_t)p * 520 + 8 + g * HD + d];
  }
  attn[i] = num / den;
}

// ---------------------------------------------------------------------------
extern "C" void kernel_launch(void* const* d_in, const int* in_sizes, int n_in,
                              void* d_out, int out_size, void* d_ws, size_t ws_size,
                              hipStream_t stream) {
  (void)in_sizes; (void)n_in; (void)out_size; (void)ws_size;
  const float* x    = (const float*)d_in[0];   // [1,1,4,4096]
  const float* wqkv = (const float*)d_in[1];   // [4096, 6144]
  const float* wo   = (const float*)d_in[2];   // [4096, 4096]
  const float* rot  = (const float*)d_in[3];   // [128, 128]
  const float* ck   = (const float*)d_in[4];   // [4, 8, 8192, 128]
  const float* cv   = (const float*)d_in[5];   // [4, 8, 8192, 128]
  const int*   pos  = (const int*)d_in[6];     // scalar
  float* out = (float*)d_out;

  float* ws    = (float*)d_ws;
  float* qkvb  = ws;                 // 4*6144                = 24576 floats
  float* qkrot = ws + 24576;         // 160*128               = 20480
  float* attnb = ws + 45056;         // 4*4096                = 16384
  float* apart = ws + 61440;         // 32*4*520              = 66560
  float* gpart = ws + 128000;        // 16*24576 (reused)     = 393216
  float* apad  = ws + 521216;        // 16*4096 padded A      = 65536

  // 1) qkv = x @ wqkv : pad A, then 384 N-tiles x 16 K-slices
  pad_a16_4096<<<dim3(256), 256, 0, stream>>>(x, apad, BB);
  gemm_a16_splitk<<<dim3(48, NSPLIT), 256, 0, stream>>>(apad, wqkv, gpart, BB, HIDDEN, NQKV);
  reduce_slices<<<dim3(96), 256, 0, stream>>>(gpart, qkvb, BB * NQKV, NSPLIT);

  // 2) rotary: [q;k] @ rot_mat (80 WMMA tiles)
  rotary_wmma<<<dim3(10), 256, 0, stream>>>(qkvb, rot, qkrot);

  // 3) split-KV flash decode attention + combine
  attn_flash_kernel<<<dim3(BB * NKVH, NPART), 256, 0, stream>>>(qkrot, qkvb, ck, cv, pos, apart);
  attn_combine_kernel<<<dim3(64), 256, 0, stream>>>(apart, attnb);

  // 4) out = attn @ wo : pad A, then 256 N-tiles x 16 K-slices
  pad_a16_4096<<<dim3(256), 256, 0, stream>>>(attnb, apad, BB);
  gemm_a16_splitk<<<dim3(32, NSPLIT), 256, 0, stream>>>(apad, wo, gpart, BB, HIDDEN, HIDDEN);
  reduce_slices<<<dim3(64), 256, 0, stream>>>(gpart, out, BB * HIDDEN, NSPLIT);
}